// HuberLoss_6820408066431
// MI455X (gfx1250) — compile-verified
//
#include <hip/hip_runtime.h>

// Huber loss (delta = 1.0), mean over all elements.
// Bandwidth-bound: 512 MiB in / 4 B out -> ~22.5 us floor @ 23.3 TB/s.
// Strategy: NT b128 streaming loads, 4x unroll w/ 4 accumulators,
// wave32 shuffle reduction, deterministic two-stage tree via d_ws.

typedef float v4f __attribute__((ext_vector_type(4)));

#define HL_BLOCK 256
#define HL_GRID  2048
#define HL_WAVES (HL_BLOCK / 32)

__device__ __forceinline__ float huber4(v4f a, v4f b) {
  float s = 0.0f;
#pragma unroll
  for (int k = 0; k < 4; ++k) {
    float d  = a[k] - b[k];
    float ad = __builtin_fabsf(d);
    float q  = __builtin_fminf(ad, 1.0f);
    // q*(ad - 0.5q): = 0.5 d^2 for ad<=1, = ad-0.5 for ad>1
    s = __builtin_fmaf(q, ad - 0.5f * q, s);
  }
  return s;
}

__global__ __launch_bounds__(HL_BLOCK) void huber_partial_kernel(
    const v4f* __restrict__ a4, const v4f* __restrict__ b4,
    float* __restrict__ partial, int n4) {
  const int tid    = blockIdx.x * HL_BLOCK + threadIdx.x;
  const int stride = gridDim.x * HL_BLOCK;

  float acc0 = 0.0f, acc1 = 0.0f, acc2 = 0.0f, acc3 = 0.0f;

  int i = tid;
  // 4x unrolled grid-stride main loop: 8 outstanding b128 NT loads / iter.
  for (; i + 3 * stride < n4; i += 4 * stride) {
    v4f a0 = __builtin_nontemporal_load(&a4[i]);
    v4f b0 = __builtin_nontemporal_load(&b4[i]);
    v4f a1 = __builtin_nontemporal_load(&a4[i + stride]);
    v4f b1 = __builtin_nontemporal_load(&b4[i + stride]);
    v4f a2 = __builtin_nontemporal_load(&a4[i + 2 * stride]);
    v4f b2 = __builtin_nontemporal_load(&b4[i + 2 * stride]);
    v4f a3 = __builtin_nontemporal_load(&a4[i + 3 * stride]);
    v4f b3 = __builtin_nontemporal_load(&b4[i + 3 * stride]);
    acc0 += huber4(a0, b0);
    acc1 += huber4(a1, b1);
    acc2 += huber4(a2, b2);
    acc3 += huber4(a3, b3);
  }
  // Remainder (n4 not necessarily a multiple of 4*stride).
  for (; i < n4; i += stride) {
    v4f a = __builtin_nontemporal_load(&a4[i]);
    v4f b = __builtin_nontemporal_load(&b4[i]);
    acc0 += huber4(a, b);
  }

  float acc = (acc0 + acc1) + (acc2 + acc3);

  // wave32 tree reduction (CDNA5 is wave32: 5 steps).
#pragma unroll
  for (int off = 16; off > 0; off >>= 1)
    acc += __shfl_down(acc, off, 32);

  __shared__ float wsum[HL_WAVES];
  if ((threadIdx.x & 31) == 0) wsum[threadIdx.x >> 5] = acc;
  __syncthreads();

  if (threadIdx.x == 0) {
    float s = 0.0f;
#pragma unroll
    for (int w = 0; w < HL_WAVES; ++w) s += wsum[w];
    partial[blockIdx.x] = s;
  }
}

// Single-block deterministic final reduction; also folds in any scalar tail
// (elements beyond the last full float4) and applies the 1/N mean scale.
__global__ __launch_bounds__(HL_BLOCK) void huber_final_kernel(
    const float* __restrict__ partial, int nparts,
    const float* __restrict__ a, const float* __restrict__ b,
    long long tail_start, long long n, float* __restrict__ out) {
  float acc = 0.0f;
  for (int i = threadIdx.x; i < nparts; i += HL_BLOCK)
    acc += partial[i];

  if (threadIdx.x == 0) {
    for (long long i = tail_start; i < n; ++i) {
      float d  = a[i] - b[i];
      float ad = __builtin_fabsf(d);
      float q  = __builtin_fminf(ad, 1.0f);
      acc = __builtin_fmaf(q, ad - 0.5f * q, acc);
    }
  }

#pragma unroll
  for (int off = 16; off > 0; off >>= 1)
    acc += __shfl_down(acc, off, 32);

  __shared__ float wsum[HL_WAVES];
  if ((threadIdx.x & 31) == 0) wsum[threadIdx.x >> 5] = acc;
  __syncthreads();

  if (threadIdx.x == 0) {
    float s = 0.0f;
#pragma unroll
    for (int w = 0; w < HL_WAVES; ++w) s += wsum[w];
    out[0] = s * (1.0f / (float)n);
  }
}

extern "C" void kernel_launch(void* const* d_in, const int* in_sizes, int n_in,
                              void* d_out, int out_size, void* d_ws, size_t ws_size,
                              hipStream_t stream) {
  const float* yhat = (const float*)d_in[0];
  const float* y    = (const float*)d_in[1];
  float* out        = (float*)d_out;
  float* partial    = (float*)d_ws;

  const long long n  = (long long)in_sizes[0];   // 16384*4096 = 67,108,864
  const int n4       = (int)(n >> 2);            // full float4 groups
  const long long tail_start = (long long)n4 << 2;

  int grid = HL_GRID;
  // Safety: never write more partials than scratch can hold.
  if ((size_t)grid * sizeof(float) > ws_size) {
    grid = (int)(ws_size / sizeof(float));
    if (grid < 1) grid = 1;
  }

  huber_partial_kernel<<<grid, HL_BLOCK, 0, stream>>>(
      (const v4f*)yhat, (const v4f*)y, partial, n4);
  huber_final_kernel<<<1, HL_BLOCK, 0, stream>>>(
      partial, grid, yhat, y, tail_start, n, out);
}